// NAO_53609781789396
// MI455X (gfx1250) — compile-verified
//
#include <hip/hip_runtime.h>
#include <hip/hip_bf16.h>

// ---------------------------------------------------------------------------
// Problem constants (from reference)
// ---------------------------------------------------------------------------
#define NGRID   32
#define NTOKEN  1024            // NGRID*NGRID
#define NTOK2   2048            // 2*NTOKEN
#define HEAD    4
#define DK      64
#define NLAYER  4
#define FDIM    64
#define BATCH   8

typedef __attribute__((ext_vector_type(2))) float v2f;
typedef __attribute__((ext_vector_type(8))) float v8f;

// ---------------------------------------------------------------------------
// WMMA helpers: V_WMMA_F32_16X16X4_F32, D(16x16,f32) = A(16x4,f32)*B(4x16,f32)+C
// A layout: lane L<16 -> row=L, {v0,v1}={K0,K1};  lane L>=16 -> row=L-16, {K2,K3}
// B layout: lane L<16 -> col=L, {v0,v1}={K0,K1};  lane L>=16 -> col=L-16, {K2,K3}
// C/D     : lane L<16 -> col=L, vr -> row r;      lane L>=16 -> col=L-16, row r+8
// ---------------------------------------------------------------------------
__device__ __forceinline__ v8f wmma4(v2f a, v2f b, v8f c) {
  return __builtin_amdgcn_wmma_f32_16x16x4_f32(
      /*neg_a=*/false, a, /*neg_b=*/false, b,
      /*c_mod=*/(short)0, c, /*reuse_a=*/false, /*reuse_b=*/false);
}

// A fragment from row-major src (row = M index, lda = leading dim)
__device__ __forceinline__ v2f frag_a(const float* src, int lda, int m0, int k0, int lane) {
  int row = m0 + (lane & 15);
  int kk  = k0 + ((lane >> 4) << 1);
  v2f a; a.x = src[row * lda + kk]; a.y = src[row * lda + kk + 1];
  return a;
}

// A fragment from column-major src: A[row,k] = src[k*lds + row]
__device__ __forceinline__ v2f frag_a_t(const float* src, int lds, int m0, int k0, int lane) {
  int row = m0 + (lane & 15);
  int kk  = k0 + ((lane >> 4) << 1);
  v2f a; a.x = src[kk * lds + row]; a.y = src[(kk + 1) * lds + row];
  return a;
}

// B fragment from row-major src (row = K index, ldb = leading dim)
__device__ __forceinline__ v2f frag_b(const float* src, int ldb, int k0, int n0, int lane) {
  int col = n0 + (lane & 15);
  int kk  = k0 + ((lane >> 4) << 1);
  v2f b; b.x = src[kk * ldb + col]; b.y = src[(kk + 1) * ldb + col];
  return b;
}

// ---------------------------------------------------------------------------
// Edge-MLP kernel: weight[n,m] = MLP_k(grid[n],grid[m]); weightf = MLP_f(...)
// MLP: 4 -> 32 (leaky) -> 64 (leaky) -> 1.   All weights staged in LDS.
// ---------------------------------------------------------------------------
#define MLPF 2337   // 128 + 32 + 2048 + 64 + 64 + 1 floats per net

__device__ __forceinline__ float edge_net(const float* s, float e0, float e1, float e2, float e3) {
  const float* W0 = s;
  const float* b0 = s + 128;
  const float* W1 = s + 160;
  const float* b1 = s + 2208;
  const float* W2 = s + 2272;
  float bias2 = s[2336];
  float h0[32];
#pragma unroll
  for (int u = 0; u < 32; ++u) {
    float t = b0[u];
    t = fmaf(W0[u * 4 + 0], e0, t);
    t = fmaf(W0[u * 4 + 1], e1, t);
    t = fmaf(W0[u * 4 + 2], e2, t);
    t = fmaf(W0[u * 4 + 3], e3, t);
    h0[u] = t > 0.f ? t : 0.01f * t;
  }
  float out = bias2;
  for (int v = 0; v < 64; ++v) {
    float t = b1[v];
    const float* w = W1 + v * 32;
#pragma unroll
    for (int u = 0; u < 32; ++u) t = fmaf(w[u], h0[u], t);
    t = t > 0.f ? t : 0.01f * t;
    out = fmaf(W2[v], t, out);
  }
  return out;
}

__global__ void edge_mlp_kernel(const float* kW0, const float* kb0, const float* kW1,
                                const float* kb1, const float* kW2, const float* kb2,
                                const float* fW0, const float* fb0, const float* fW1,
                                const float* fb1, const float* fW2, const float* fb2,
                                float* wout, float* fout) {
  __shared__ float sh[2 * MLPF];
  float* sk = sh;
  float* sf = sh + MLPF;
  for (int i = threadIdx.x; i < MLPF; i += blockDim.x) {
    float v, vf;
    if (i < 128)       { v = kW0[i];        vf = fW0[i]; }
    else if (i < 160)  { v = kb0[i - 128];  vf = fb0[i - 128]; }
    else if (i < 2208) { v = kW1[i - 160];  vf = fW1[i - 160]; }
    else if (i < 2272) { v = kb1[i - 2208]; vf = fb1[i - 2208]; }
    else if (i < 2336) { v = kW2[i - 2272]; vf = fW2[i - 2272]; }
    else               { v = kb2[0];        vf = fb2[0]; }
    sk[i] = v; sf[i] = vf;
  }
  __syncthreads();
  int idx = blockIdx.x * blockDim.x + threadIdx.x;   // 0 .. 1024*1024-1
  int n = idx >> 10, m = idx & 1023;
  const float DXc = 2.0f / 31.0f;
  float e0 = -1.f + DXc * (float)(n >> 5);
  float e1 = -1.f + DXc * (float)(n & 31);
  float e2 = -1.f + DXc * (float)(m >> 5);
  float e3 = -1.f + DXc * (float)(m & 31);
  wout[idx] = edge_net(sk, e0, e1, e2, e3);
  fout[idx] = edge_net(sf, e0, e1, e2, e3);
}

// ---------------------------------------------------------------------------
// Global layernorm over whole (2048,64) slab per batch (axes (-2,-1))
// ---------------------------------------------------------------------------
__global__ void ln0_reduce_kernel(const float* xy, float* stats) {
  int b = blockIdx.x;
  const float* p = xy + (size_t)b * NTOK2 * FDIM;
  const int N = NTOK2 * FDIM;   // 131072
  float s = 0.f, ss = 0.f;
  for (int i = threadIdx.x; i < N; i += 256) {
    float v = p[i];
    s += v; ss += v * v;
  }
  __shared__ float sh[512];
  sh[threadIdx.x] = s; sh[256 + threadIdx.x] = ss;
  __syncthreads();
  for (int st = 128; st >= 1; st >>= 1) {
    if ((int)threadIdx.x < st) {
      sh[threadIdx.x]       += sh[threadIdx.x + st];
      sh[256 + threadIdx.x] += sh[256 + threadIdx.x + st];
    }
    __syncthreads();
  }
  if (threadIdx.x == 0) {
    float mean = sh[0] / (float)N;
    float var  = sh[256] / (float)N - mean * mean;
    stats[2 * b]     = mean;
    stats[2 * b + 1] = rsqrtf(var + 1e-5f);
  }
}

__global__ void ln0_apply_kernel(const float* xy, const float* g, const float* bb,
                                 const float* stats, float* V) {
  int idx = blockIdx.x * blockDim.x + threadIdx.x;   // over 8*2048*64
  int b = idx >> 17;             // / (2048*64)
  int r = idx & 131071;
  V[idx] = (xy[idx] - stats[2 * b]) * stats[2 * b + 1] * g[r] + bb[r];
}

// ---------------------------------------------------------------------------
// Q/K projection:  Out[b,h,n,k] = Vin[b,n,:] . W[h][k][:] + bias[h][k]
// grid.x: 0..31 -> Q(b*4+h), 32..63 -> K(b*4+h); grid.y: 32 (strip of 4 row tiles)
// block: 128 threads = 4 waves; each wave owns one 16-row M tile and all 4
// 16-col N tiles (1x4 register blocking: A-frag reused by 4 WMMAs).
// ---------------------------------------------------------------------------
__global__ void qk_kernel(const float* Vin, const float* Wq, const float* bq,
                          const float* Wk, const float* bk, float* Q, float* K) {
  int gx = blockIdx.x;
  bool isK = gx >= 32;
  int bh = gx & 31;
  const float* W    = isK ? Wk : Wq;
  const float* bias = isK ? bk : bq;
  float* Out        = isK ? K  : Q;
  int h = bh & 3;
  int b = bh >> 2;
  const float* A     = Vin + (size_t)b * NTOK2 * FDIM;
  const float* Wh    = W + h * DK * FDIM;
  const float* biash = bias + h * DK;
  int wave = threadIdx.x >> 5, lane = threadIdx.x & 31;
  int m0 = blockIdx.y * 64 + wave * 16;
  v8f c[4] = {};
  for (int k0 = 0; k0 < FDIM; k0 += 4) {
    v2f a = frag_a(A, FDIM, m0, k0, lane);
    int kk = k0 + ((lane >> 4) << 1);
#pragma unroll
    for (int t = 0; t < 4; ++t) {
      int col = t * 16 + (lane & 15);
      v2f bf; bf.x = Wh[col * FDIM + kk]; bf.y = Wh[col * FDIM + kk + 1];
      c[t] = wmma4(a, bf, c[t]);
    }
  }
  float* dst = Out + (size_t)bh * NTOK2 * DK;
  int rbase = m0 + ((lane >> 4) << 3);
#pragma unroll
  for (int t = 0; t < 4; ++t) {
    int col = t * 16 + (lane & 15);
    float bv = biash[col];
#pragma unroll
    for (int r = 0; r < 8; ++r) dst[(rbase + r) * DK + col] = c[t][r] + bv;
  }
}

// ---------------------------------------------------------------------------
// M_h = s * K_hᵀ @ B   (64x64 per (b,h)); B = Vin_b (kdim=2048) or xy_b (kdim=1024)
// 32 blocks (one per b,h) x 128 threads; wave -> one 16-row (k1) tile,
// all 4 f tiles held in registers (A-frag reused 4x per K step).
// ---------------------------------------------------------------------------
__global__ void ktv_kernel(const float* Kbuf, const float* Bsrc, int bstride,
                           int kdim, float s, float* Mout) {
  int bh = blockIdx.x;
  int b = bh >> 2;
  int wave = threadIdx.x >> 5, lane = threadIdx.x & 31;
  int m0 = wave * 16;          // k1 tile
  const float* Kh = Kbuf + (size_t)bh * NTOK2 * DK;
  const float* B  = Bsrc + (size_t)b * bstride;
  v8f c[4] = {};
  for (int k0 = 0; k0 < kdim; k0 += 4) {
    v2f a = frag_a_t(Kh, DK, m0, k0, lane);    // A[k1, m] = Kh[m*DK + k1]
#pragma unroll
    for (int t = 0; t < 4; ++t) {
      v2f bf = frag_b(B, FDIM, k0, t * 16, lane);   // B[m, f]
      c[t] = wmma4(a, bf, c[t]);
    }
  }
  float* dst = Mout + (size_t)bh * DK * FDIM;
  int rbase = m0 + ((lane >> 4) << 3);
#pragma unroll
  for (int t = 0; t < 4; ++t) {
    int col = t * 16 + (lane & 15);
#pragma unroll
    for (int r = 0; r < 8; ++r) dst[(rbase + r) * FDIM + col] = c[t][r] * s;
  }
}

// ---------------------------------------------------------------------------
// Out[b,n,f] = sum_h Q[b,h,n,:] @ M[b,h,:,:]   (used for mid and for S)
// grid: (8, 32 row strips), block 128 (wave -> 16-row tile, 4 f tiles in regs)
// ---------------------------------------------------------------------------
__global__ void qm_kernel(const float* Q, const float* Mb, float* Out) {
  int b = blockIdx.x;
  int wave = threadIdx.x >> 5, lane = threadIdx.x & 31;
  int m0 = blockIdx.y * 64 + wave * 16;
  v8f c[4] = {};
  for (int h = 0; h < HEAD; ++h) {
    const float* Qh = Q + (size_t)(b * HEAD + h) * NTOK2 * DK;
    const float* Mh = Mb + (size_t)(b * HEAD + h) * DK * FDIM;
    for (int k0 = 0; k0 < DK; k0 += 4) {
      v2f a = frag_a(Qh, DK, m0, k0, lane);
#pragma unroll
      for (int t = 0; t < 4; ++t) {
        v2f bf = frag_b(Mh, FDIM, k0, t * 16, lane);
        c[t] = wmma4(a, bf, c[t]);
      }
    }
  }
  float* dst = Out + (size_t)b * NTOK2 * FDIM;
  int rbase = m0 + ((lane >> 4) << 3);
#pragma unroll
  for (int t = 0; t < 4; ++t) {
    int col = t * 16 + (lane & 15);
#pragma unroll
    for (int r = 0; r < 8; ++r) dst[(rbase + r) * FDIM + col] = c[t][r];
  }
}

// ---------------------------------------------------------------------------
// Per-row layernorm (axis -1, 64 wide) + residual: Vout = LN(mid)*g+b + Vin
// one wave per row, 2 elements per lane, shfl_xor reduction (wave32)
// ---------------------------------------------------------------------------
__global__ void ln_res_kernel(const float* mid, const float* vin, const float* g,
                              const float* bb, float* vout) {
  int row = blockIdx.x * 8 + (threadIdx.x >> 5);
  int lane = threadIdx.x & 31;
  const float* mr = mid + (size_t)row * FDIM;
  float x0 = mr[lane], x1 = mr[lane + 32];
  float s = x0 + x1, ss = x0 * x0 + x1 * x1;
#pragma unroll
  for (int m = 16; m >= 1; m >>= 1) {
    s  += __shfl_xor(s, m, 32);
    ss += __shfl_xor(ss, m, 32);
  }
  float mean = s * (1.f / 64.f);
  float var  = ss * (1.f / 64.f) - mean * mean;
  float inv  = rsqrtf(var + 1e-5f);
  size_t base = (size_t)row * FDIM;
  vout[base + lane]      = (x0 - mean) * inv * g[lane]      + bb[lane]      + vin[base + lane];
  vout[base + lane + 32] = (x1 - mean) * inv * g[lane + 32] + bb[lane + 32] + vin[base + lane + 32];
}

// ---------------------------------------------------------------------------
// Final: out[b,j,f] = mul * sum_{k<2048} A[j,k] * S[b,k,f]
//   A[j,k] = weight[k,j] (k<1024) or weightf[k-1024,j]   (strided loads!)
// grid: (8, 16 j-strips), block 128 (wave -> 16-row j tile, 4 f tiles in regs)
// Strided A fragment reused by 4 WMMAs; prefetch of A strip 16 K-steps ahead.
// ---------------------------------------------------------------------------
__global__ void final_kernel(const float* weight, const float* weightf,
                             const float* S, float* out, float mul) {
  int b = blockIdx.x;
  int wave = threadIdx.x >> 5, lane = threadIdx.x & 31;
  int m0 = blockIdx.y * 64 + wave * 16;   // j tile
  const float* Sb = S + (size_t)b * NTOK2 * FDIM;
  int row = m0 + (lane & 15);
  v8f c[4] = {};
  for (int k0 = 0; k0 < NTOK2; k0 += 4) {
    const float* W = (k0 < NTOKEN) ? weight : weightf;
    int kb = (k0 & (NTOKEN - 1)) + ((lane >> 4) << 1);
    v2f a; a.x = W[kb * NTOKEN + row]; a.y = W[(kb + 1) * NTOKEN + row];
    if ((k0 & 60) == 0 && (k0 & (NTOKEN - 1)) + 64 < NTOKEN) {
      __builtin_prefetch(&W[(kb + 64) * NTOKEN + row], 0, 3);
    }
#pragma unroll
    for (int t = 0; t < 4; ++t) {
      v2f bf = frag_b(Sb, FDIM, k0, t * 16, lane);
      c[t] = wmma4(a, bf, c[t]);
    }
  }
  float* dst = out + (size_t)b * NTOKEN * FDIM;
  int rbase = m0 + ((lane >> 4) << 3);
#pragma unroll
  for (int t = 0; t < 4; ++t) {
    int col = t * 16 + (lane & 15);
#pragma unroll
    for (int r = 0; r < 8; ++r) dst[(rbase + r) * FDIM + col] = c[t][r] * mul;
  }
}

// ---------------------------------------------------------------------------
// Host-side orchestration
// ---------------------------------------------------------------------------
extern "C" void kernel_launch(void* const* d_in, const int* in_sizes, int n_in,
                              void* d_out, int out_size, void* d_ws, size_t ws_size,
                              hipStream_t stream) {
  (void)in_sizes; (void)n_in; (void)out_size; (void)ws_size;
  const float* xy   = (const float*)d_in[0];
  const float* kW0  = (const float*)d_in[1];
  const float* kb0  = (const float*)d_in[2];
  const float* kW1  = (const float*)d_in[3];
  const float* kb1  = (const float*)d_in[4];
  const float* kW2  = (const float*)d_in[5];
  const float* kb2  = (const float*)d_in[6];
  const float* fW0  = (const float*)d_in[7];
  const float* fb0  = (const float*)d_in[8];
  const float* fW1  = (const float*)d_in[9];
  const float* fb1  = (const float*)d_in[10];
  const float* fW2  = (const float*)d_in[11];
  const float* fb2  = (const float*)d_in[12];
  const float* Wq   = (const float*)d_in[13];
  const float* bq   = (const float*)d_in[14];
  const float* Wk   = (const float*)d_in[15];
  const float* bk   = (const float*)d_in[16];
  const float* ln0g = (const float*)d_in[17];
  const float* ln0b = (const float*)d_in[18];
  const float* lng  = (const float*)d_in[19];
  const float* lnb  = (const float*)d_in[20];
  float* out = (float*)d_out;

  // workspace layout (floats)
  float* ws = (float*)d_ws;
  const size_t SZ_1M = 1048576;                    // 1024*1024
  float* w_weight  = ws;                           // 1M
  float* w_weightf = ws + SZ_1M;                   // 1M
  float* V0        = ws + 2 * SZ_1M;               // 8*2048*64 = 1M
  float* V1        = ws + 3 * SZ_1M;               // 1M
  float* Qb        = ws + 4 * SZ_1M;               // 8*4*2048*64 = 4M
  float* Kb        = ws + 8 * SZ_1M;               // 4M
  float* Mb        = ws + 12 * SZ_1M;              // 8*4*64*64 = 128K
  float* MID       = ws + 12 * SZ_1M + 131072;     // 1M (mid / S)
  float* stats     = ws + 13 * SZ_1M + 131072;     // 16

  const float DXc   = 2.0f / 31.0f;
  const float dxdy  = DXc * DXc;
  const float scale = 0.125f;   // 1/sqrt(64)

  // 1. edge MLPs -> weight, weightf
  edge_mlp_kernel<<<4096, 256, 0, stream>>>(kW0, kb0, kW1, kb1, kW2, kb2,
                                            fW0, fb0, fW1, fb1, fW2, fb2,
                                            w_weight, w_weightf);
  // 2. global layernorm -> V0
  ln0_reduce_kernel<<<BATCH, 256, 0, stream>>>(xy, stats);
  ln0_apply_kernel<<<4096, 256, 0, stream>>>(xy, ln0g, ln0b, stats, V0);

  float* Va = V0;
  float* Vb = V1;
  // 3. layers 0..2:  mid = dx^2*scale * sum_h Q_h (K_hᵀ Vin);  Vin' = LN(mid)+Vin
  for (int j = 0; j < NLAYER - 1; ++j) {
    qk_kernel<<<dim3(64, NTOK2 / 64), 128, 0, stream>>>(
        Va, Wq + (size_t)j * HEAD * DK * FDIM, bq + (size_t)j * HEAD * DK,
        Wk + (size_t)j * HEAD * DK * FDIM, bk + (size_t)j * HEAD * DK, Qb, Kb);
    ktv_kernel<<<32, 128, 0, stream>>>(Kb, Va, NTOK2 * FDIM, NTOK2,
                                       scale * dxdy, Mb);
    qm_kernel<<<dim3(BATCH, NTOK2 / 64), 128, 0, stream>>>(Qb, Mb, MID);
    ln_res_kernel<<<2048, 256, 0, stream>>>(MID, Va, lng + j * FDIM,
                                            lnb + j * FDIM, Vb);
    float* t = Va; Va = Vb; Vb = t;
  }

  // 4. last layer:  N_h = scale * K_h[:1024]ᵀ x_u ;  S = sum_h Q_h N_h
  {
    int j = NLAYER - 1;
    qk_kernel<<<dim3(64, NTOK2 / 64), 128, 0, stream>>>(
        Va, Wq + (size_t)j * HEAD * DK * FDIM, bq + (size_t)j * HEAD * DK,
        Wk + (size_t)j * HEAD * DK * FDIM, bk + (size_t)j * HEAD * DK, Qb, Kb);
    ktv_kernel<<<32, 128, 0, stream>>>(Kb, xy, NTOK2 * FDIM, NTOKEN, scale, Mb);
    qm_kernel<<<dim3(BATCH, NTOK2 / 64), 128, 0, stream>>>(Qb, Mb, MID);  // S
    // 5. out[b,j,f] = dx^4 * ( weightᵀ Su + weightfᵀ Sf )
    final_kernel<<<dim3(BATCH, NTOKEN / 16 / 4), 128, 0, stream>>>(
        w_weight, w_weightf, MID, out, dxdy * dxdy);
  }
}